// FreqSM_23991687315890
// MI455X (gfx1250) — compile-verified
//
#include <hip/hip_runtime.h>

#define TWO_PI 6.28318530717958647692f

typedef float v2f __attribute__((ext_vector_type(2)));
typedef float v8f __attribute__((ext_vector_type(8)));

static __device__ __forceinline__ float dev_sigmoid(float x) { return 1.f / (1.f + __expf(-x)); }
static __device__ __forceinline__ float dev_silu(float x) { return x / (1.f + __expf(-x)); }

// ---------------------------------------------------------------------------
// fp32 WMMA GEMM:  C[M,N] = A[M,K] @ B[K,N] (+bias), row-major, K%4==0.
// One wave computes a 16(M) x 64(N) tile: 4 accumulators, 4 WMMAs per K-step
// sharing a single v2f A fragment (global_load_b64).
// Out-of-range rows/cols are CLAMPED (not masked) so all loads are branchless;
// padded rows of A only affect padded rows of D, padded cols of B only padded
// cols of D, and those are masked at the store.
// grid = (ceil(N/64), ceil(M/16), batch), block = 32 (one wave32).
// A/B advance by (bz % aMod/bMod)*stride so shared weights broadcast.
// biasMode: 0 none, 1 per-column (N), 2 per-row (M).
// ---------------------------------------------------------------------------
__global__ void fsm_gemm(const float* __restrict__ A, const float* __restrict__ Bm,
                         float* __restrict__ C, const float* __restrict__ bias,
                         int M, int N, int K, int lda, int ldb, int ldc,
                         long sA, long sB, long sC, int aMod, int bMod, int biasMode) {
  int bz = blockIdx.z;
  A  += (long)(bz % aMod) * sA;
  Bm += (long)(bz % bMod) * sB;
  C  += (long)bz * sC;
  const int lane = threadIdx.x & 31;
  const int half = lane >> 4;          // 0: K pair {0,1}, 1: K pair {2,3}
  const int lid  = lane & 15;
  const int row0 = blockIdx.y * 16;
  const int col0 = blockIdx.x * 64;

  int ar = row0 + lid;
  if (ar >= M) ar = M - 1;                       // clamp -> branchless loads
  const float* Ap = A + (long)ar * lda + half * 2;

  int c0 = col0 + lid, c1 = c0 + 16, c2 = c0 + 32, c3 = c0 + 48;
  int k0c = c0 >= N ? N - 1 : c0;
  int k1c = c1 >= N ? N - 1 : c1;
  int k2c = c2 >= N ? N - 1 : c2;
  int k3c = c3 >= N ? N - 1 : c3;
  const long boff = (long)(half * 2) * ldb;
  const float* B0 = Bm + boff + k0c;
  const float* B1 = Bm + boff + k1c;
  const float* B2 = Bm + boff + k2c;
  const float* B3 = Bm + boff + k3c;
  const long bstep = 4L * ldb;

  v8f ac0 = {0.f, 0.f, 0.f, 0.f, 0.f, 0.f, 0.f, 0.f};
  v8f ac1 = ac0, ac2 = ac0, ac3 = ac0;

  for (int kk = 0; kk < K; kk += 4) {
    v2f a = *(const v2f*)Ap;
    Ap += 4;
    v2f b;
    b.x = B0[0]; b.y = B0[ldb]; B0 += bstep;
    ac0 = __builtin_amdgcn_wmma_f32_16x16x4_f32(false, a, false, b, (short)0, ac0, false, false);
    b.x = B1[0]; b.y = B1[ldb]; B1 += bstep;
    ac1 = __builtin_amdgcn_wmma_f32_16x16x4_f32(false, a, false, b, (short)0, ac1, false, false);
    b.x = B2[0]; b.y = B2[ldb]; B2 += bstep;
    ac2 = __builtin_amdgcn_wmma_f32_16x16x4_f32(false, a, false, b, (short)0, ac2, false, false);
    b.x = B3[0]; b.y = B3[ldb]; B3 += bstep;
    ac3 = __builtin_amdgcn_wmma_f32_16x16x4_f32(false, a, false, b, (short)0, ac3, false, false);
  }

  auto store_tile = [&](v8f acc, int cc) {
    if (cc >= N) return;
    float bv = (biasMode == 1) ? bias[cc] : 0.f;
#pragma unroll
    for (int v = 0; v < 8; ++v) {
      int r = row0 + v + half * 8;
      if (r >= M) continue;
      float val = acc[v] + bv;
      if (biasMode == 2) val += bias[r];
      C[(long)r * ldc + cc] = val;
    }
  };
  store_tile(ac0, c0);
  store_tile(ac1, c1);
  store_tile(ac2, c2);
  store_tile(ac3, c3);
}

// ---------------------------------------------------------------------------
// LayerNorm over last dim of contiguous rows. One wave per row. block (32, 8).
// ---------------------------------------------------------------------------
__global__ void fsm_ln_rows(const float* __restrict__ in, const float* __restrict__ g,
                            const float* __restrict__ b, float* __restrict__ out,
                            int rows, int C, float eps) {
  int row = blockIdx.x * blockDim.y + threadIdx.y;
  if (row >= rows) return;
  const float* p = in + (long)row * C;
  int lane = threadIdx.x;
  float s = 0.f, s2 = 0.f;
  for (int c = lane; c < C; c += 32) { float v = p[c]; s += v; s2 += v * v; }
  for (int m = 16; m; m >>= 1) { s += __shfl_xor(s, m, 32); s2 += __shfl_xor(s2, m, 32); }
  float mean = s / C;
  float inv = rsqrtf(s2 / C - mean * mean + eps);
  float* q = out + (long)row * C;
  for (int c = lane; c < C; c += 32) q[c] = (p[c] - mean) * inv * g[c] + b[c];
}

// LayerNorm over channel for (strided) NCHW input; output contiguous NCHW.
__global__ void fsm_ln_nchw(const float* __restrict__ in, const float* __restrict__ g,
                            const float* __restrict__ b, float* __restrict__ out,
                            int Bn, int C, int HW, long sb_in, float eps) {
  int idx = blockIdx.x * blockDim.y + threadIdx.y;   // over Bn*HW pixels
  if (idx >= Bn * HW) return;
  int bb = idx / HW, p = idx % HW;
  const float* base = in + (long)bb * sb_in + p;
  int lane = threadIdx.x;
  float s = 0.f, s2 = 0.f;
  for (int c = lane; c < C; c += 32) { float v = base[(long)c * HW]; s += v; s2 += v * v; }
  for (int m = 16; m; m >>= 1) { s += __shfl_xor(s, m, 32); s2 += __shfl_xor(s2, m, 32); }
  float mean = s / C;
  float inv = rsqrtf(s2 / C - mean * mean + eps);
  for (int c = lane; c < C; c += 32)
    out[((long)bb * C + c) * HW + p] = (base[(long)c * HW] - mean) * inv * g[c] + b[c];
}

// LN(out_norm) then multiply by silu(z) pulled from the in_proj output.
__global__ void fsm_ln_silu_mul(const float* __restrict__ y, const float* __restrict__ g,
                                const float* __restrict__ b, const float* __restrict__ xz,
                                int zld, int zoff, float* __restrict__ out,
                                int rows, int C, float eps) {
  int row = blockIdx.x * blockDim.y + threadIdx.y;
  if (row >= rows) return;
  const float* p = y + (long)row * C;
  int lane = threadIdx.x;
  float s = 0.f, s2 = 0.f;
  for (int c = lane; c < C; c += 32) { float v = p[c]; s += v; s2 += v * v; }
  for (int m = 16; m; m >>= 1) { s += __shfl_xor(s, m, 32); s2 += __shfl_xor(s2, m, 32); }
  float mean = s / C;
  float inv = rsqrtf(s2 / C - mean * mean + eps);
  float* q = out + (long)row * C;
  const float* zp = xz + (long)row * zld + zoff;
  for (int c = lane; c < C; c += 32) {
    float ln = (p[c] - mean) * inv * g[c] + b[c];
    q[c] = ln * dev_silu(zp[c]);
  }
}

// ---------------------------------------------------------------------------
// Depthwise 3x3 SAME conv with arbitrary input strides; output contiguous NCHW.
// act==1 -> SiLU fused.
// ---------------------------------------------------------------------------
__global__ void fsm_dwconv3(const float* __restrict__ in, const float* __restrict__ w,
                            const float* __restrict__ bias, float* __restrict__ out,
                            int Bn, int C, int H, int Wd,
                            long sb, long sc, long sh, long sw, int act) {
  long idx = (long)blockIdx.x * blockDim.x + threadIdx.x;
  long total = (long)Bn * C * H * Wd;
  if (idx >= total) return;
  int wv = idx % Wd;
  int h  = (idx / Wd) % H;
  int c  = (idx / ((long)Wd * H)) % C;
  int b  = idx / ((long)Wd * H * C);
  const float* ip = in + (long)b * sb + (long)c * sc;
  const float* wp = w + (long)c * 9;
  float acc = bias[c];
#pragma unroll
  for (int kh = -1; kh <= 1; ++kh) {
    int hh = h + kh;
    if (hh < 0 || hh >= H) continue;
#pragma unroll
    for (int kw = -1; kw <= 1; ++kw) {
      int ww = wv + kw;
      if (ww < 0 || ww >= Wd) continue;
      acc += ip[(long)hh * sh + (long)ww * sw] * wp[(kh + 1) * 3 + (kw + 1)];
    }
  }
  if (act) acc = dev_silu(acc);
  out[idx] = acc;
}

// Build the 4-direction scan inputs xs (B,4,di,L) from NCHW xc.
__global__ void fsm_build_xs(const float* __restrict__ xc, float* __restrict__ xs,
                             int Bn, int di, int H, int Wd) {
  long L = (long)H * Wd;
  long idx = (long)blockIdx.x * blockDim.x + threadIdx.x;
  long total = (long)Bn * di * L;
  if (idx >= total) return;
  long l = idx % L;
  int d = (idx / L) % di;
  int b = idx / (L * di);
  const float* cp = xc + ((long)b * di + d) * L;
  float v0 = cp[l];
  int h2 = l % H, w2 = l / H;              // x_wh[l = w*H + h] = xc[h, w]
  float v1 = cp[(long)h2 * Wd + w2];
  long base = ((long)b * 4 * di + d) * L;
  xs[base + l] = v0;
  xs[base + (long)di * L + l] = v1;
  xs[base + 2L * di * L + (L - 1 - l)] = v0;
  xs[base + 3L * di * L + (L - 1 - l)] = v1;
}

// dts = softplus(dts + dt_b[k,d])
__global__ void fsm_softplus(float* __restrict__ dts, const float* __restrict__ dtb,
                             int di, long L, long n) {
  long idx = (long)blockIdx.x * blockDim.x + threadIdx.x;
  if (idx >= n) return;
  int d = (idx / L) % di;
  int k = (idx / (L * di)) & 3;
  float v = dts[idx] + dtb[k * di + d];
  dts[idx] = (v > 20.f) ? v : log1pf(__expf(v));
}

// ---------------------------------------------------------------------------
// Selective scan: grid = B*4 blocks, block = di threads (one per channel).
// h[n] = h[n]*exp(dt*A[n]) + dt*u*Bs[n,t];  y = sum_n h[n]*Cs[n,t] + u*D.
// Bs/Cs (shared across the block) staged through LDS in 64-step chunks.
// ---------------------------------------------------------------------------
__global__ void fsm_scan(const float* __restrict__ xs, const float* __restrict__ dts,
                         const float* __restrict__ xdbl, const float* __restrict__ A_log,
                         const float* __restrict__ Dp, float* __restrict__ outy,
                         int di, int L, int R, int nC) {
  int bk = blockIdx.x;
  int k = bk & 3;
  int d = threadIdx.x;
  const float* xsp = xs  + ((long)bk * di + d) * L;
  const float* dtp = dts + ((long)bk * di + d) * L;
  const float* Bp  = xdbl + ((long)bk * nC + R) * L;
  const float* Cp  = xdbl + ((long)bk * nC + R + 16) * L;
  float* op = outy + ((long)bk * di + d) * L;
  float a[16], h[16];
#pragma unroll
  for (int n = 0; n < 16; ++n) {
    a[n] = -__expf(A_log[((long)k * di + d) * 16 + n]);
    h[n] = 0.f;
  }
  float Dv = Dp[k * di + d];
  __shared__ float sB[16 * 64];
  __shared__ float sC[16 * 64];
  for (int t0 = 0; t0 < L; t0 += 64) {
    __syncthreads();
    for (int e = threadIdx.x; e < 16 * 64; e += blockDim.x) {
      int n = e >> 6, tt = e & 63;
      sB[e] = Bp[(long)n * L + t0 + tt];
      sC[e] = Cp[(long)n * L + t0 + tt];
    }
    __syncthreads();
    if (t0 + 64 < L) {
      __builtin_prefetch(&dtp[t0 + 64], 0, 1);
      __builtin_prefetch(&xsp[t0 + 64], 0, 1);
    }
    for (int tt = 0; tt < 64; ++tt) {
      float dt = dtp[t0 + tt];
      float u  = xsp[t0 + tt];
      float du = dt * u;
      float y = 0.f;
#pragma unroll
      for (int n = 0; n < 16; ++n) {
        h[n] = h[n] * __expf(dt * a[n]) + du * sB[n * 64 + tt];
        y += h[n] * sC[n * 64 + tt];
      }
      op[t0 + tt] = y + u * Dv;
    }
  }
}

// Merge 4 scan directions (with flips / HW-transpose) -> NHWC tokens x di.
__global__ void fsm_merge(const float* __restrict__ oy, float* __restrict__ ym,
                          int Bn, int di, int H, int Wd) {
  long L = (long)H * Wd;
  long idx = (long)blockIdx.x * blockDim.x + threadIdx.x;
  long total = (long)Bn * L * di;
  if (idx >= total) return;
  int d = idx % di;
  long l = (idx / di) % L;
  int b = idx / (di * L);
  int h = l / Wd, w = l % Wd;
  long lw = (long)w * H + h;
  const float* p0 = oy + ((long)(b * 4 + 0) * di + d) * L;
  const float* p1 = oy + ((long)(b * 4 + 1) * di + d) * L;
  const float* p2 = oy + ((long)(b * 4 + 2) * di + d) * L;
  const float* p3 = oy + ((long)(b * 4 + 3) * di + d) * L;
  ym[((long)b * L + l) * di + d] = p0[l] + p2[L - 1 - l] + p1[lw] + p3[L - 1 - lw];
}

// ---------------------------------------------------------------------------
// DFT kernels (ortho FFT via twiddle tables).
// ---------------------------------------------------------------------------
__global__ void fsm_fft_tables(float* tWc, float* tWs, float* tHc, float* tHs) {
  int i = blockIdx.x * blockDim.x + threadIdx.x;
  if (i < 64 * 33) {
    int w = i / 33, k = i % 33;
    float ang = TWO_PI * (float)((w * k) % 64) / 64.f;
    tWc[i] = cosf(ang); tWs[i] = sinf(ang);
  }
  if (i < 64 * 64) {
    int h = i / 64, k = i % 64;
    float ang = TWO_PI * (float)((h * k) % 64) / 64.f;
    tHc[i] = cosf(ang); tHs[i] = sinf(ang);
  }
}

// real->complex DFT along W. Input strides (3 outer dims); output (o,kw,i) contig.
__global__ void fsm_dft_r2c(const float* __restrict__ in, float* __restrict__ Re,
                            float* __restrict__ Im, const float* __restrict__ tc,
                            const float* __restrict__ ts,
                            int d1, int d2, int d3, long s1, long s2, long s3,
                            long isw, int W, int Wr, int ninner) {
  long total = (long)d1 * d2 * d3 * Wr * ninner;
  long idx = (long)blockIdx.x * blockDim.x + threadIdx.x;
  if (idx >= total) return;
  int i = idx % ninner;
  long t = idx / ninner;
  int kw = t % Wr;
  long o = t / Wr;
  int i3 = o % d3; long o2 = o / d3;
  int i2 = o2 % d2; int i1 = o2 / d2;
  const float* p = in + (long)i1 * s1 + (long)i2 * s2 + (long)i3 * s3 + i;
  float sr = 0.f, si = 0.f;
  for (int w = 0; w < W; ++w) {
    float v = p[(long)w * isw];
    sr += v * tc[w * Wr + kw];
    si -= v * ts[w * Wr + kw];
  }
  Re[idx] = sr; Im[idx] = si;
}

// complex->complex DFT along H; in/out layout (o, h, j) contiguous.
// sign=+1 forward (e^{-i}), sign=-1 inverse (e^{+i}).
__global__ void fsm_dft_c2c(const float* __restrict__ Ri, const float* __restrict__ Ii,
                            float* __restrict__ Ro, float* __restrict__ Io,
                            const float* __restrict__ tc, const float* __restrict__ ts,
                            int nouter, int H, long ninner, float sign, float scale) {
  long total = (long)nouter * H * ninner;
  long idx = (long)blockIdx.x * blockDim.x + threadIdx.x;
  if (idx >= total) return;
  long i = idx % ninner;
  int k = (idx / ninner) % H;
  long o = idx / (ninner * H);
  long base = o * H * ninner + i;
  float sr = 0.f, si = 0.f;
  for (int h = 0; h < H; ++h) {
    float re = Ri[base + (long)h * ninner];
    float im = Ii[base + (long)h * ninner];
    float c = tc[h * H + k], s = ts[h * H + k];
    sr += re * c + sign * im * s;
    si += im * c - sign * re * s;
  }
  Ro[idx] = sr * scale; Io[idx] = si * scale;
}

// complex->real inverse DFT along W (Hermitian fold, alpha weights), strided out.
__global__ void fsm_dft_c2r(const float* __restrict__ Re, const float* __restrict__ Im,
                            float* __restrict__ out, const float* __restrict__ resid,
                            const float* __restrict__ tc, const float* __restrict__ ts,
                            int d1, int d2, int d3, long s1, long s2, long s3,
                            int W, int Wr, int ninner, long osw, float scale) {
  long total = (long)d1 * d2 * d3 * W * ninner;
  long idx = (long)blockIdx.x * blockDim.x + threadIdx.x;
  if (idx >= total) return;
  int i = idx % ninner;
  long t = idx / ninner;
  int w = t % W;
  long o = t / W;
  int i3 = o % d3; long o2 = o / d3;
  int i2 = o2 % d2; int i1 = o2 / d2;
  long ib = o * (long)Wr * ninner + i;
  float acc = 0.f;
  for (int kw = 0; kw < Wr; ++kw) {
    float alpha = (kw == 0 || kw == Wr - 1) ? 1.f : 2.f;
    acc += alpha * (Re[ib + (long)kw * ninner] * tc[w * Wr + kw] -
                    Im[ib + (long)kw * ninner] * ts[w * Wr + kw]);
  }
  long oo = (long)i1 * s1 + (long)i2 * s2 + (long)i3 * s3 + (long)w * osw + i;
  float v = acc * scale;
  if (resid) v += resid[oo];
  out[oo] = v;
}

// Interleave Re/Im -> [..., 2c] / [..., 2c+1] channel pairs (and inverse).
__global__ void fsm_pack_ri(const float* __restrict__ Re, const float* __restrict__ Im,
                            float* __restrict__ xr, long n, int C) {
  long idx = (long)blockIdx.x * blockDim.x + threadIdx.x;
  if (idx >= n) return;
  int c = idx % C; long rest = idx / C;
  xr[rest * 2 * C + 2 * c] = Re[idx];
  xr[rest * 2 * C + 2 * c + 1] = Im[idx];
}
__global__ void fsm_unpack_ri(const float* __restrict__ att, float* __restrict__ Re,
                              float* __restrict__ Im, long n, int C) {
  long idx = (long)blockIdx.x * blockDim.x + threadIdx.x;
  if (idx >= n) return;
  int c = idx % C; long rest = idx / C;
  Re[idx] = att[rest * 2 * C + 2 * c];
  Im[idx] = att[rest * 2 * C + 2 * c + 1];
}

// freq_bg: mask * bilinear-resized complex weight, applied in place to Yr/Yi.
static __device__ __forceinline__ float fsm_bilin16(const float* img, float ys, float xs) {
  int y0 = (int)floorf(ys); y0 = y0 < 0 ? 0 : (y0 > 15 ? 15 : y0);
  int x0 = (int)floorf(xs); x0 = x0 < 0 ? 0 : (x0 > 15 ? 15 : x0);
  int y1 = y0 + 1 > 15 ? 15 : y0 + 1;
  int x1 = x0 + 1 > 15 ? 15 : x0 + 1;
  float wy = ys - (float)y0, wx = xs - (float)x0;
  float a = img[y0 * 16 + x0] * (1.f - wx) + img[y0 * 16 + x1] * wx;
  float b = img[y1 * 16 + x0] * (1.f - wx) + img[y1 * 16 + x1] * wx;
  return a * (1.f - wy) + b * wy;
}
__global__ void fsm_maskw(float* __restrict__ Yr, float* __restrict__ Yi,
                          const float* __restrict__ cw, int Bn, int Cb,
                          float radius, int keep_low) {
  long total = (long)Bn * Cb * 64 * 33;
  long idx = (long)blockIdx.x * blockDim.x + threadIdx.x;
  if (idx >= total) return;
  int kw = idx % 33;
  int kh = (idx / 33) % 64;
  int cc = (idx / (33 * 64)) % Cb;
  float dy = (float)(kh - 32), dx = (float)(kw - 16);
  float dist = sqrtf(dy * dy + dx * dx);
  float mask = keep_low ? (dist > radius ? 0.f : 1.f)
                        : (dist < radius ? 0.f : 1.f);
  float ys = kh * (15.f / 63.f);
  float xs = kw * (15.f / 32.f);
  float wr = fsm_bilin16(cw + (long)cc * 256, ys, xs);
  float wi = fsm_bilin16(cw + ((long)Cb + cc) * 256, ys, xs);
  float yr = Yr[idx], yi = Yi[idx];
  Yr[idx] = (yr * wr - yi * wi) * mask;
  Yi[idx] = (yr * wi + yi * wr) * mask;
}

// ---------------------------------------------------------------------------
// Small elementwise / reduction helpers.
// ---------------------------------------------------------------------------
__global__ void fsm_add(const float* a, const float* b, float* o, long n) {
  long i = (long)blockIdx.x * blockDim.x + threadIdx.x;
  if (i < n) o[i] = a[i] + b[i];
}
// out NHWC[(b*HW+p)*C + c] = in[b*sb + c*HW + p]
__global__ void fsm_nhwc_from_nchw(const float* in, float* out, int Bn, int C, int HW, long sb) {
  long n = (long)Bn * HW * C;
  long i = (long)blockIdx.x * blockDim.x + threadIdx.x;
  if (i >= n) return;
  int c = i % C; long p = (i / C) % HW; int b = i / ((long)C * HW);
  out[i] = in[(long)b * sb + (long)c * HW + p];
}
// X1N (NCHW) = NHWC(xspa) + NHWC(xfrq) + x1_orig(slice of x)
__global__ void fsm_combine_x1(const float* xspa, const float* xfrq, const float* x,
                               float* out, int Bn, int C, int HW, long sbx) {
  long n = (long)Bn * C * HW;
  long i = (long)blockIdx.x * blockDim.x + threadIdx.x;
  if (i >= n) return;
  long p = i % HW; int c = (i / HW) % C; int b = i / ((long)HW * C);
  long nh = ((long)b * HW + p) * C + c;
  out[i] = xspa[nh] + xfrq[nh] + x[(long)b * sbx + (long)c * HW + p];
}
// ECA: global avg pool (wave per (b,c))
__global__ void fsm_pool(const float* x, float* y, int n_bc, int HW) {
  int bc = blockIdx.x * blockDim.y + threadIdx.y;
  if (bc >= n_bc) return;
  const float* p = x + (long)bc * HW;
  float s = 0.f;
  for (int i = threadIdx.x; i < HW; i += 32) s += p[i];
  for (int m = 16; m; m >>= 1) s += __shfl_xor(s, m, 32);
  if (threadIdx.x == 0) y[bc] = s / (float)HW;
}
__global__ void fsm_eca_gate(const float* y, const float* w, float* gate, int Bn, int C) {
  int idx = blockIdx.x * blockDim.x + threadIdx.x;
  if (idx >= Bn * C) return;
  int c = idx % C;
  float ym1 = (c > 0) ? y[idx - 1] : 0.f;
  float yp1 = (c < C - 1) ? y[idx + 1] : 0.f;
  gate[idx] = dev_sigmoid(ym1 * w[0] + y[idx] * w[1] + yp1 * w[2]);
}
__global__ void fsm_scale_bc(float* x, const float* gate, long n, int HW) {
  long i = (long)blockIdx.x * blockDim.x + threadIdx.x;
  if (i < n) x[i] *= gate[i / HW];
}
// Per-channel mean/var over (B, HW): one block per channel.
__global__ void fsm_chan_stats(const float* y, float* mv, int Bn, int C, int HW) {
  int c = blockIdx.x;
  __shared__ float ss[256], s2[256];
  float a = 0.f, b2 = 0.f;
  int n = Bn * HW;
  for (int i = threadIdx.x; i < n; i += 256) {
    int bb = i / HW, p = i % HW;
    float v = y[((long)bb * C + c) * HW + p];
    a += v; b2 += v * v;
  }
  ss[threadIdx.x] = a; s2[threadIdx.x] = b2;
  __syncthreads();
  for (int st = 128; st; st >>= 1) {
    if (threadIdx.x < st) { ss[threadIdx.x] += ss[threadIdx.x + st]; s2[threadIdx.x] += s2[threadIdx.x + st]; }
    __syncthreads();
  }
  if (threadIdx.x == 0) {
    float mean = ss[0] / n;
    mv[c] = mean;
    mv[C + c] = s2[0] / n - mean * mean;
  }
}
__global__ void fsm_bn_relu_add(const float* y, const float* mv, const float* g,
                                const float* b, const float* resid, long rsb,
                                float* out, int Bn, int C, int HW) {
  long n = (long)Bn * C * HW;
  long i = (long)blockIdx.x * blockDim.x + threadIdx.x;
  if (i >= n) return;
  long p = i % HW; int c = (i / HW) % C; int bb = i / ((long)HW * C);
  float v = (y[i] - mv[c]) * rsqrtf(mv[C + c] + 1e-5f) * g[c] + b[c];
  v = v > 0.f ? v : 0.f;
  out[i] = v + resid[(long)bb * rsb + (long)c * HW + p];
}
// cat2[(b*Ctot + c0 + c)*HW + p] = eb[i] + xi[i]
__global__ void fsm_res_to_cat(const float* eb, const float* xi, float* cat2,
                               int Bn, int C, int HW, int c0, int Ctot) {
  long n = (long)Bn * C * HW;
  long i = (long)blockIdx.x * blockDim.x + threadIdx.x;
  if (i >= n) return;
  long p = i % HW; int c = (i / HW) % C; int b = i / ((long)HW * C);
  cat2[((long)b * Ctot + c0 + c) * HW + p] = eb[i] + xi[i];
}

// ---------------------------------------------------------------------------
// Host launch
// ---------------------------------------------------------------------------
static inline int nblk(long n, int t) { return (int)((n + t - 1) / t); }

extern "C" void kernel_launch(void* const* d_in, const int* in_sizes, int n_in,
                              void* d_out, int out_size, void* d_ws, size_t ws_size,
                              hipStream_t stream) {
  (void)in_sizes; (void)n_in; (void)out_size; (void)ws_size;
  enum { Bn = 4, Hh = 64, Wdm = 64, HWp = 4096, T1 = 16384, WrK = 33,
         T2 = 8448, L1 = 4096, L2 = 2112, DI1 = 128, DI2 = 256 };
#define IN(i) ((const float*)d_in[i])
  const float* x = IN(0);
  const float* v_n1g = IN(1); const float* v_n1b = IN(2);
  const float* v_n2g = IN(3); const float* v_n2b = IN(4);
  const float* sp_lng = IN(5); const float* sp_lnb = IN(6);
  // spa ss2d (d=64, di=128, R=4)
  const float* s1_inw = IN(7);  const float* s1_inb = IN(8);
  const float* s1_cw  = IN(9);  const float* s1_cb  = IN(10);
  const float* s1_xpw = IN(11); const float* s1_dtw = IN(12);
  const float* s1_dtb = IN(13); const float* s1_alog = IN(14);
  const float* s1_D   = IN(15); const float* s1_ong = IN(16);
  const float* s1_onb = IN(17); const float* s1_opw = IN(18);
  const float* s1_opb = IN(19);
  // frq ss2d (d=128, di=256, R=8)
  const float* s2_inw = IN(20); const float* s2_inb = IN(21);
  const float* s2_cw  = IN(22); const float* s2_cb  = IN(23);
  const float* s2_xpw = IN(24); const float* s2_dtw = IN(25);
  const float* s2_dtb = IN(26); const float* s2_alog = IN(27);
  const float* s2_D   = IN(28); const float* s2_ong = IN(29);
  const float* s2_onb = IN(30); const float* s2_opw = IN(31);
  const float* s2_opb = IN(32);
  // freq_bg
  const float* fb_n1g = IN(33); const float* fb_n1b = IN(34);
  const float* fb_wh  = IN(35); const float* fb_wl  = IN(36);
  const float* fb_pw  = IN(37); const float* fb_pb  = IN(38);
  const float* fb_bg  = IN(39); const float* fb_bb  = IN(40);
  // top level
  const float* t_n1g = IN(41); const float* t_n1b = IN(42);
  const float* eca_w = IN(43);
  const float* dw_w  = IN(44); const float* dw_b = IN(45);
  const float* pw_w  = IN(46); const float* pw_b = IN(47);
  const float* prj_w = IN(48); const float* prj_b = IN(49);
  float* out = (float*)d_out;

  float* W = (float*)d_ws;
  size_t off = 0;
  auto alloc = [&](long n) { float* p = W + off; off += (size_t)n; return p; };
  float* tWc = alloc(64 * 33); float* tWs = alloc(64 * 33);
  float* tHc = alloc(64 * 64); float* tHs = alloc(64 * 64);
  float* XP   = alloc((long)T1 * 64);
  float* LN1  = alloc((long)T1 * 64);
  float* LN2  = alloc((long)T1 * 64);
  float* FZ   = alloc((long)T1 * 64);
  float* XSPA = alloc((long)T1 * 64);
  float* XFRQ = alloc((long)T1 * 64);
  float* X1N  = alloc((long)Bn * 64 * HWp);
  float* X2N  = alloc((long)Bn * 64 * HWp);
  float* XZ   = alloc((long)T2 * 512);        // >= T1*256 too
  float* XC   = alloc((long)Bn * 256 * L2);
  float* XS   = alloc((long)Bn * 4 * 256 * L2);
  float* XDBL = alloc((long)Bn * 4 * 36 * L1);
  float* DTS  = alloc((long)Bn * 4 * 256 * L2);
  float* OUTY = alloc((long)Bn * 4 * 256 * L2);
  float* YM   = alloc((long)T2 * 256);
  float* YA   = alloc((long)T2 * 256);
  float* SOUT = alloc((long)T2 * 128);
  float* RE1  = alloc((long)T2 * 64);
  float* IM1  = alloc((long)T2 * 64);
  float* AR   = alloc((long)T2 * 64);
  float* AI   = alloc((long)T2 * 64);
  float* XR   = alloc((long)T2 * 128);
  float* LNB  = alloc((long)Bn * 64 * HWp);
  float* FBR  = alloc((long)Bn * 51 * 64 * 33);
  float* FBI  = alloc((long)Bn * 51 * 64 * 33);
  float* FBR2 = alloc((long)Bn * 51 * 64 * 33);
  float* FBI2 = alloc((long)Bn * 51 * 64 * 33);
  float* CAT  = alloc((long)Bn * 64 * HWp);
  float* YB   = alloc((long)Bn * 64 * HWp);
  float* MV   = alloc(128);
  float* TBuf = alloc((long)Bn * 64 * HWp);
  float* DBuf = alloc((long)Bn * 64 * HWp);
  float* PB   = alloc(Bn * 64);
  float* GB   = alloc(Bn * 64);
  float* CAT2 = alloc((long)Bn * 128 * HWp);

  dim3 w32(32, 8);

  // twiddle tables
  fsm_fft_tables<<<nblk(4096, 256), 256, 0, stream>>>(tWc, tWs, tHc, tHs);

  // xp = NHWC(x1)
  fsm_nhwc_from_nchw<<<nblk((long)T1 * 64, 256), 256, 0, stream>>>(x, XP, Bn, 64, HWp,
                                                                   (long)128 * HWp);
  // LN1 = LN(xp, vss.n1); LN2 = LN(LN1, spa.ln, 1e-6)
  fsm_ln_rows<<<nblk(T1, 8), w32, 0, stream>>>(XP, v_n1g, v_n1b, LN1, T1, 64, 1e-5f);
  fsm_ln_rows<<<nblk(T1, 8), w32, 0, stream>>>(LN1, sp_lng, sp_lnb, LN2, T1, 64, 1e-6f);

  // ---------------- ss2d (spa: d=64, di=128, H=64, W=64, L=4096) ----------------
  fsm_gemm<<<dim3(4, 1024, 1), 32, 0, stream>>>(LN2, s1_inw, XZ, s1_inb,
      T1, 256, 64, 64, 256, 256, 0, 0, 0, 1, 1, 1);
  fsm_dwconv3<<<nblk((long)Bn * DI1 * HWp, 256), 256, 0, stream>>>(XZ, s1_cw, s1_cb, XC,
      Bn, DI1, 64, 64, (long)HWp * 256, 1L, 64L * 256, 256L, 1);
  fsm_build_xs<<<nblk((long)Bn * DI1 * L1, 256), 256, 0, stream>>>(XC, XS, Bn, DI1, 64, 64);
  fsm_gemm<<<dim3(64, 3, 16), 32, 0, stream>>>(s1_xpw, XS, XDBL, nullptr,
      36, L1, DI1, DI1, L1, L1, 36L * DI1, (long)DI1 * L1, 36L * L1, 4, 16, 0);
  fsm_gemm<<<dim3(64, 8, 16), 32, 0, stream>>>(s1_dtw, XDBL, DTS, nullptr,
      DI1, L1, 4, 4, L1, L1, (long)DI1 * 4, 36L * L1, (long)DI1 * L1, 4, 16, 0);
  fsm_softplus<<<nblk(16L * DI1 * L1, 256), 256, 0, stream>>>(DTS, s1_dtb, DI1, L1,
      16L * DI1 * L1);
  fsm_scan<<<16, DI1, 0, stream>>>(XS, DTS, XDBL, s1_alog, s1_D, OUTY, DI1, L1, 4, 36);
  fsm_merge<<<nblk((long)Bn * L1 * DI1, 256), 256, 0, stream>>>(OUTY, YM, Bn, DI1, 64, 64);
  fsm_ln_silu_mul<<<nblk(T1, 8), w32, 0, stream>>>(YM, s1_ong, s1_onb, XZ, 256, 128,
      YA, T1, DI1, 1e-5f);
  fsm_gemm<<<dim3(1, 1024, 1), 32, 0, stream>>>(YA, s1_opw, SOUT, s1_opb,
      T1, 64, DI1, DI1, 64, 64, 0, 0, 0, 1, 1, 1);
  fsm_add<<<nblk((long)T1 * 64, 256), 256, 0, stream>>>(SOUT, LN1, XSPA, (long)T1 * 64);

  // ---------------- freq_vss_block ----------------
  fsm_ln_rows<<<nblk(T1, 8), w32, 0, stream>>>(XP, v_n2g, v_n2b, FZ, T1, 64, 1e-5f);
  fsm_dft_r2c<<<nblk((long)T2 * 64, 256), 256, 0, stream>>>(FZ, RE1, IM1, tWc, tWs,
      Bn * 64, 1, 1, 4096, 0, 0, 64, 64, 33, 64);
  fsm_dft_c2c<<<nblk((long)T2 * 64, 256), 256, 0, stream>>>(RE1, IM1, AR, AI, tHc, tHs,
      Bn, 64, 2112, 1.f, 1.f / 64.f);
  fsm_pack_ri<<<nblk((long)T2 * 64, 256), 256, 0, stream>>>(AR, AI, XR, (long)T2 * 64, 64);

  // ---------------- ss2d (frq: d=128, di=256, H=64, W=33, L=2112) ----------------
  fsm_gemm<<<dim3(8, 528, 1), 32, 0, stream>>>(XR, s2_inw, XZ, s2_inb,
      T2, 512, 128, 128, 512, 512, 0, 0, 0, 1, 1, 1);
  fsm_dwconv3<<<nblk((long)Bn * DI2 * L2, 256), 256, 0, stream>>>(XZ, s2_cw, s2_cb, XC,
      Bn, DI2, 64, 33, (long)L2 * 512, 1L, 33L * 512, 512L, 1);
  fsm_build_xs<<<nblk((long)Bn * DI2 * L2, 256), 256, 0, stream>>>(XC, XS, Bn, DI2, 64, 33);
  fsm_gemm<<<dim3(33, 3, 16), 32, 0, stream>>>(s2_xpw, XS, XDBL, nullptr,
      40, L2, DI2, DI2, L2, L2, 40L * DI2, (long)DI2 * L2, 40L * L2, 4, 16, 0);
  fsm_gemm<<<dim3(33, 16, 16), 32, 0, stream>>>(s2_dtw, XDBL, DTS, nullptr,
      DI2, L2, 8, 8, L2, L2, (long)DI2 * 8, 40L * L2, (long)DI2 * L2, 4, 16, 0);
  fsm_softplus<<<nblk(16L * DI2 * L2, 256), 256, 0, stream>>>(DTS, s2_dtb, DI2, L2,
      16L * DI2 * L2);
  fsm_scan<<<16, DI2, 0, stream>>>(XS, DTS, XDBL, s2_alog, s2_D, OUTY, DI2, L2, 8, 40);
  fsm_merge<<<nblk((long)Bn * L2 * DI2, 256), 256, 0, stream>>>(OUTY, YM, Bn, DI2, 64, 33);
  fsm_ln_silu_mul<<<nblk(T2, 8), w32, 0, stream>>>(YM, s2_ong, s2_onb, XZ, 512, 256,
      YA, T2, DI2, 1e-5f);
  fsm_gemm<<<dim3(2, 528, 1), 32, 0, stream>>>(YA, s2_opw, SOUT, s2_opb,
      T2, 128, DI2, DI2, 128, 128, 0, 0, 0, 1, 1, 1);

  // irfft2 + residual (FZ)
  fsm_unpack_ri<<<nblk((long)T2 * 64, 256), 256, 0, stream>>>(SOUT, AR, AI, (long)T2 * 64, 64);
  fsm_dft_c2c<<<nblk((long)T2 * 64, 256), 256, 0, stream>>>(AR, AI, RE1, IM1, tHc, tHs,
      Bn, 64, 2112, -1.f, 1.f);
  fsm_dft_c2r<<<nblk((long)T1 * 64, 256), 256, 0, stream>>>(RE1, IM1, XFRQ, FZ, tWc, tWs,
      Bn * 64, 1, 1, 4096, 0, 0, 64, 33, 64, 64, 1.f / 64.f);

  // x1_new = NCHW(xspa + xfrq) + x1
  fsm_combine_x1<<<nblk((long)Bn * 64 * HWp, 256), 256, 0, stream>>>(XSPA, XFRQ, x, X1N,
      Bn, 64, HWp, (long)128 * HWp);

  // ---------------- freq_bg on x2 ----------------
  fsm_ln_nchw<<<nblk(Bn * HWp, 8), w32, 0, stream>>>(x + (long)64 * HWp, fb_n1g, fb_n1b,
      LNB, Bn, 64, HWp, (long)128 * HWp, 1e-5f);
  auto fb_branch = [&](int c0, int Cb, const float* cw, int keep_low) {
    long n2 = (long)Bn * Cb * 64 * 33;
    fsm_dft_r2c<<<nblk(n2, 256), 256, 0, stream>>>(LNB + (long)c0 * HWp, FBR, FBI,
        tWc, tWs, Bn, Cb, 64, (long)64 * HWp, (long)HWp, 64L, 1, 64, 33, 1);
    fsm_dft_c2c<<<nblk(n2, 256), 256, 0, stream>>>(FBR, FBI, FBR2, FBI2, tHc, tHs,
        Bn * Cb, 64, 33, 1.f, 1.f / 64.f);
    fsm_maskw<<<nblk(n2, 256), 256, 0, stream>>>(FBR2, FBI2, cw, Bn, Cb, 8.f, keep_low);
    fsm_dft_c2c<<<nblk(n2, 256), 256, 0, stream>>>(FBR2, FBI2, FBR, FBI, tHc, tHs,
        Bn * Cb, 64, 33, -1.f, 1.f);
    fsm_dft_c2r<<<nblk((long)Bn * Cb * HWp, 256), 256, 0, stream>>>(FBR, FBI,
        CAT + (long)c0 * HWp, nullptr, tWc, tWs,
        Bn, Cb, 64, (long)64 * HWp, (long)HWp, 64L, 64, 33, 1, 1, 1.f / 64.f);
  };
  fb_branch(0, 51, fb_wh, 0);
  fb_branch(51, 13, fb_wl, 1);
  fsm_gemm<<<dim3(64, 4, Bn), 32, 0, stream>>>(fb_pw, CAT, YB, fb_pb,
      64, HWp, 64, 64, HWp, HWp, 0, (long)64 * HWp, (long)64 * HWp, 1, Bn, 2);
  fsm_chan_stats<<<64, 256, 0, stream>>>(YB, MV, Bn, 64, HWp);
  fsm_bn_relu_add<<<nblk((long)Bn * 64 * HWp, 256), 256, 0, stream>>>(YB, MV, fb_bg, fb_bb,
      x + (long)64 * HWp, (long)128 * HWp, X2N, Bn, 64, HWp);

  // ---------------- res branches -> CAT2 ----------------
  auto res_branch = [&](const float* Xi, int c0) {
    fsm_ln_nchw<<<nblk(Bn * HWp, 8), w32, 0, stream>>>(Xi, t_n1g, t_n1b, TBuf,
        Bn, 64, HWp, (long)64 * HWp, 1e-5f);
    fsm_pool<<<nblk(Bn * 64, 8), w32, 0, stream>>>(TBuf, PB, Bn * 64, HWp);
    fsm_eca_gate<<<1, 256, 0, stream>>>(PB, eca_w, GB, Bn, 64);
    fsm_scale_bc<<<nblk((long)Bn * 64 * HWp, 256), 256, 0, stream>>>(TBuf, GB,
        (long)Bn * 64 * HWp, HWp);
    fsm_dwconv3<<<nblk((long)Bn * 64 * HWp, 256), 256, 0, stream>>>(TBuf, dw_w, dw_b, DBuf,
        Bn, 64, 64, 64, (long)64 * HWp, (long)HWp, 64L, 1L, 0);
    fsm_gemm<<<dim3(64, 4, Bn), 32, 0, stream>>>(pw_w, DBuf, SOUT, pw_b,
        64, HWp, 64, 64, HWp, HWp, 0, (long)64 * HWp, (long)64 * HWp, 1, Bn, 2);
    fsm_res_to_cat<<<nblk((long)Bn * 64 * HWp, 256), 256, 0, stream>>>(SOUT, Xi, CAT2,
        Bn, 64, HWp, c0, 128);
  };
  res_branch(X1N, 0);
  res_branch(X2N, 64);

  // ---------------- final conv1x1 -> out ----------------
  fsm_gemm<<<dim3(64, 8, Bn), 32, 0, stream>>>(prj_w, CAT2, out, prj_b,
      128, HWp, 128, 128, HWp, HWp, 0, (long)128 * HWp, (long)128 * HWp, 1, Bn, 2);
#undef IN
}